// BaseTokenizingNet_66726611910955
// MI455X (gfx1250) — compile-verified
//
#include <hip/hip_runtime.h>

// Problem constants (from the reference)
#define N_TOKENS   65536
#define N_FEAT     8
#define VOCAB      1026
#define EMB        256            // floats per embedding row
#define EMB4       (EMB / 4)      // 64 float4 chunks per row

// Native vector type so __builtin_nontemporal_store is legal and lowers to
// global_store_b128 with the NT temporal hint (CDNA5 TH field).
typedef float v4f __attribute__((ext_vector_type(4)));

// One 64-thread block (2 wave32) per token.
//  - indices address is block-uniform -> scalar s_load (8 ints, one 32B line)
//  - each lane loads one float4 of each of the 8 gathered rows -> 8x
//    fully-coalesced global_load_b128 (1024B per row across 64 lanes)
//  - sum in feature order, store with non-temporal hint so the 64MB output
//    stream does not evict the 8.4MB L2-resident embedding tables.
__global__ __launch_bounds__(64) void emb_gather_sum_kernel(
    const int* __restrict__ idx,        // [N_TOKENS][N_FEAT] int32
    const v4f* __restrict__ tab,        // [N_FEAT * VOCAB][EMB4] float4 view
    v4f* __restrict__ out)              // [N_TOKENS][EMB4] float4 view
{
    const int token = blockIdx.x;       // 0 .. N_TOKENS-1
    const int lane  = threadIdx.x;      // 0 .. 63, owns 4 floats of the row

    // Block-uniform index fetch -> compiler scalarizes to s_load.
    const int b  = token * N_FEAT;
    const int i0 = idx[b + 0];
    const int i1 = idx[b + 1];
    const int i2 = idx[b + 2];
    const int i3 = idx[b + 3];
    const int i4 = idx[b + 4];
    const int i5 = idx[b + 5];
    const int i6 = idx[b + 6];
    const int i7 = idx[b + 7];

    // Row f, vocab index i lives at float4 offset (f*VOCAB + i)*EMB4.
    // Issue all 8 independent b128 loads before consuming any (latency hiding).
    const v4f a0 = tab[(0 * VOCAB + i0) * EMB4 + lane];
    const v4f a1 = tab[(1 * VOCAB + i1) * EMB4 + lane];
    const v4f a2 = tab[(2 * VOCAB + i2) * EMB4 + lane];
    const v4f a3 = tab[(3 * VOCAB + i3) * EMB4 + lane];
    const v4f a4 = tab[(4 * VOCAB + i4) * EMB4 + lane];
    const v4f a5 = tab[(5 * VOCAB + i5) * EMB4 + lane];
    const v4f a6 = tab[(6 * VOCAB + i6) * EMB4 + lane];
    const v4f a7 = tab[(7 * VOCAB + i7) * EMB4 + lane];

    // Left-fold in feature order (matches reference reduction order).
    v4f s = a0 + a1;
    s = s + a2;
    s = s + a3;
    s = s + a4;
    s = s + a5;
    s = s + a6;
    s = s + a7;

    // Non-temporal streaming store: output is write-once, never re-read.
    __builtin_nontemporal_store(s, &out[token * EMB4 + lane]);
}

extern "C" void kernel_launch(void* const* d_in, const int* in_sizes, int n_in,
                              void* d_out, int out_size, void* d_ws, size_t ws_size,
                              hipStream_t stream)
{
    (void)in_sizes; (void)n_in; (void)out_size; (void)d_ws; (void)ws_size;

    const int*  indices = (const int*)d_in[0];          // (65536, 8) int32
    const v4f*  tables  = (const v4f*)d_in[1];          // (8, 1026, 256) f32
    v4f*        out     = (v4f*)d_out;                  // (1, 65536, 256) f32

    dim3 grid(N_TOKENS);
    dim3 block(64);
    hipLaunchKernelGGL(emb_gather_sum_kernel, grid, block, 0, stream,
                       indices, tables, out);
}